// DRKNN_76433237999915
// MI455X (gfx1250) — compile-verified
//
#include <hip/hip_runtime.h>

// ---------- types ----------
typedef float v2f  __attribute__((ext_vector_type(2)));
typedef float v8f  __attribute__((ext_vector_type(8)));
typedef unsigned int u32x4 __attribute__((ext_vector_type(4)));
typedef int   i32x8 __attribute__((ext_vector_type(8)));
typedef int   i32x4 __attribute__((ext_vector_type(4)));

// ---------- problem constants ----------
#define N_PTS   100000
#define QN      1024
#define FDIM    128
#define KNN     3
#define NSPLIT  32          // N partitioned across gridDim.y
#define CH      64          // data rows staged per chunk (CH*FDIM*4 = 32KB LDS)
#define WAVES   4
#define TPB     (WAVES*32)
#define BIGF    1.0e30f

// ---------- workspace layout (float elements) ----------
#define X2_OFF   0                                   // N_PTS floats
#define Q2_OFF   100352                              // QN floats
#define CAND_D   101376                              // QN*NSPLIT*KNN floats
#define CAND_I   (CAND_D + QN*NSPLIT*KNN)            // QN*NSPLIT*KNN ints

// =====================================================================
// Kernel 1: squared norms of data rows and query rows
// =====================================================================
__global__ __launch_bounds__(256) void drknn_norms(const float* __restrict__ X,
                                                   const float* __restrict__ Qf,
                                                   float* __restrict__ ws) {
  int i = blockIdx.x * blockDim.x + threadIdx.x;
  if (i < N_PTS) {
    const float4* r = (const float4*)(X + (size_t)i * FDIM);
    float s = 0.f;
#pragma unroll
    for (int j = 0; j < FDIM / 4; ++j) {
      float4 v = r[j];
      s += v.x * v.x + v.y * v.y + v.z * v.z + v.w * v.w;
    }
    ws[X2_OFF + i] = s;
  }
  if (i < QN) {
    const float4* r = (const float4*)(Qf + (size_t)i * FDIM);
    float s = 0.f;
#pragma unroll
    for (int j = 0; j < FDIM / 4; ++j) {
      float4 v = r[j];
      s += v.x * v.x + v.y * v.y + v.z * v.z + v.w * v.w;
    }
    ws[Q2_OFF + i] = s;
  }
}

// =====================================================================
// TDM: 2D tile load (CH rows x FDIM f32) global -> LDS, zero-fill OOB rows.
// D# packing per CDNA5 ISA ch.8 (group0 128b, group1 256b, groups2/3 zero).
// clang-23 toolchain: 6-arg builtin (g0, g1, g2, g3, g_extra, cpol).
// =====================================================================
__device__ __forceinline__ void tdm_load_chunk(unsigned lds_off,
                                               const float* gsrc,
                                               unsigned rows_avail) {
  unsigned long long ga = (unsigned long long)(const void*)gsrc;
  u32x4 g0;
  g0[0] = 1u;                                        // count=1, user mode
  g0[1] = lds_off;                                   // lds_addr (bytes)
  g0[2] = (unsigned)(ga & 0xffffffffu);              // global_addr[31:0]
  g0[3] = (unsigned)((ga >> 32) & 0x01ffffffu)       // global_addr[56:32]
        | (2u << 30);                                // type = 2 ("image")
  i32x8 g1;
  g1[0] = 0x00020000;                                // wg_mask=0, data_size=4B
  g1[1] = (int)(FDIM << 16);                         // tensor_dim0 lo16 @bits63:48
  g1[2] = (int)((rows_avail & 0xffffu) << 16);       // tensor_dim1 lo16 @bits95:80
  g1[3] = (int)(((rows_avail >> 16) & 0xffffu)       // tensor_dim1 hi16
        | ((unsigned)FDIM << 16));                   // tile_dim0 = 128
  g1[4] = (int)(CH & 0xffff);                        // tile_dim1 = CH, tile_dim2 = 0
  g1[5] = FDIM;                                      // tensor_dim0_stride lo32
  g1[6] = 0;
  g1[7] = 0;
  i32x4 g2 = {0, 0, 0, 0};
  i32x4 g3 = {0, 0, 0, 0};
  i32x8 g4 = {0, 0, 0, 0, 0, 0, 0, 0};
  __builtin_amdgcn_tensor_load_to_lds(g0, g1, g2, g3, g4, 0);
}

// =====================================================================
// Kernel 2: distance tiles via WMMA f32 16x16x4 + running top-3 per query
// grid = (QN/16, NSPLIT), block = 128 threads (4 waves)
// Double-buffered TDM pipeline: chunk c+1 streams while chunk c computes.
// =====================================================================
__global__ __launch_bounds__(TPB) void drknn_main(const float* __restrict__ X,
                                                  const float* __restrict__ Qf,
                                                  float* __restrict__ ws) {
  __shared__ float xbuf[2][CH * FDIM];       // 2 x 32 KB staged X chunks
  __shared__ float dtile[WAVES][16 * 16];    // slow-path distance dump
  __shared__ float t3d[WAVES][16][KNN];      // per-wave top-3 dists
  __shared__ int   t3i[WAVES][16][KNN];      // per-wave top-3 indices

  const int lane = threadIdx.x & 31;
  const int wave = threadIdx.x >> 5;
  const int mrow = lane & 15;                // M (A) or N (B) lane index
  const int khalf = (lane >> 4) << 1;        // K offset within 4-wide K tile
  const int q0 = blockIdx.x * 16;
  const int split = blockIdx.y;
  const int per = N_PTS / NSPLIT;            // 3125 exactly
  const int nBeg = split * per;
  const int nEnd = nBeg + per;

  // ---- A fragments (16 queries x 128 feats) held in registers ----
  v2f a[FDIM / 4];
  const float* qrow = Qf + (size_t)(q0 + mrow) * FDIM + khalf;
#pragma unroll
  for (int k = 0; k < FDIM / 4; ++k) a[k] = *(const v2f*)(qrow + 4 * k);

  // per-row query norms for the 8 rows this lane's C-VGPRs cover
  float q2r[8];
#pragma unroll
  for (int r = 0; r < 8; ++r) q2r[r] = ws[Q2_OFF + q0 + r + 8 * (lane >> 4)];

  // init per-wave top-3 and thresholds
  if (lane < 16) {
#pragma unroll
    for (int k = 0; k < KNN; ++k) { t3d[wave][lane][k] = BIGF; t3i[wave][lane][k] = 0x7fffffff; }
  }
  float thr[8];
#pragma unroll
  for (int r = 0; r < 8; ++r) thr[r] = BIGF;

  const int nch = (per + CH - 1) / CH;       // 49 chunks per split
  const unsigned ldsBase0 = (unsigned)(unsigned long long)(void*)&xbuf[0][0];
  const unsigned ldsBase1 = (unsigned)(unsigned long long)(void*)&xbuf[1][0];

  // prime the pipeline: chunk 0 -> buffer 0
  if (wave == 0) {
    tdm_load_chunk(ldsBase0, X + (size_t)nBeg * FDIM, (unsigned)(N_PTS - nBeg));
  }

  for (int c = 0; c < nch; ++c) {
    const int row0 = nBeg + c * CH;
    if (wave == 0) {
      if (c + 1 < nch) {                     // prefetch next chunk, other buffer
        const int rowN = nBeg + (c + 1) * CH;
        tdm_load_chunk((c + 1) & 1 ? ldsBase1 : ldsBase0,
                       X + (size_t)rowN * FDIM, (unsigned)(N_PTS - rowN));
        __builtin_amdgcn_s_wait_tensorcnt(1);   // in-order: chunk c complete
      } else {
        __builtin_amdgcn_s_wait_tensorcnt(0);
      }
    }
    __syncthreads();                         // chunk c visible to all waves

    const int tn0 = wave * 16;               // local row base of my tile
    v8f ce = {0.f, 0.f, 0.f, 0.f, 0.f, 0.f, 0.f, 0.f};
    v8f co = {0.f, 0.f, 0.f, 0.f, 0.f, 0.f, 0.f, 0.f};
    const float* bbase = &xbuf[c & 1][(tn0 + mrow) * FDIM + khalf];
#pragma unroll
    for (int k = 0; k < FDIM / 4; k += 2) {  // two accumulators: break RAW chain
      v2f b0 = *(const v2f*)(bbase + 4 * k);
      v2f b1 = *(const v2f*)(bbase + 4 * k + 4);
      ce = __builtin_amdgcn_wmma_f32_16x16x4_f32(
          false, a[k], false, b0, (short)0, ce, false, false);
      co = __builtin_amdgcn_wmma_f32_16x16x4_f32(
          false, a[k + 1], false, b1, (short)0, co, false, false);
    }
    v8f cacc = ce + co;

    // distances + threshold fast path
    const int ncol = row0 + tn0 + mrow;      // global data index of my column
    const int nclamp = ncol < N_PTS ? ncol : N_PTS - 1;
    const float x2c = ws[X2_OFF + nclamp];
    const bool colValid = ncol < nEnd;
    float d[8];
    bool anyHit = false;
#pragma unroll
    for (int r = 0; r < 8; ++r) {
      float dd = q2r[r] + x2c - 2.0f * cacc[r];
      dd = colValid ? dd : BIGF;
      d[r] = dd;
      anyHit |= (dd < thr[r]);
    }

    if (__any(anyHit ? 1 : 0)) {             // rare after warm-up
#pragma unroll
      for (int r = 0; r < 8; ++r)
        dtile[wave][(r + 8 * (lane >> 4)) * 16 + mrow] = d[r];
      // same-wave LDS ops are in-order (DScnt): no barrier needed
      if (lane < 16) {
        float d0 = t3d[wave][lane][0], d1 = t3d[wave][lane][1], d2 = t3d[wave][lane][2];
        int   i0 = t3i[wave][lane][0], i1 = t3i[wave][lane][1], i2 = t3i[wave][lane][2];
        const int nbase = row0 + tn0;
#pragma unroll
        for (int col = 0; col < 16; ++col) {
          float dd = dtile[wave][lane * 16 + col];
          int nn = nbase + col;
          if (dd < d2) {
            if (dd < d1) {
              d2 = d1; i2 = i1;
              if (dd < d0) { d1 = d0; i1 = i0; d0 = dd; i0 = nn; }
              else         { d1 = dd; i1 = nn; }
            } else { d2 = dd; i2 = nn; }
          }
        }
        t3d[wave][lane][0] = d0; t3d[wave][lane][1] = d1; t3d[wave][lane][2] = d2;
        t3i[wave][lane][0] = i0; t3i[wave][lane][1] = i1; t3i[wave][lane][2] = i2;
      }
#pragma unroll
      for (int r = 0; r < 8; ++r) thr[r] = t3d[wave][r + 8 * (lane >> 4)][KNN - 1];
    }
    __syncthreads();   // everyone done reading buf[c&1] before chunk c+2 reuses it
  }

  // ---- merge the 4 waves' top-3 lists, emit candidates ----
  __syncthreads();
  if (wave == 0 && lane < 16) {
    float bd0 = BIGF, bd1 = BIGF, bd2 = BIGF;
    int   bi0 = 0x7fffffff, bi1 = 0x7fffffff, bi2 = 0x7fffffff;
#pragma unroll
    for (int w = 0; w < WAVES; ++w)
#pragma unroll
      for (int k = 0; k < KNN; ++k) {
        float dd = t3d[w][lane][k];
        int nn = t3i[w][lane][k];
        if (dd < bd2 || (dd == bd2 && nn < bi2)) {
          if (dd < bd1 || (dd == bd1 && nn < bi1)) {
            bd2 = bd1; bi2 = bi1;
            if (dd < bd0 || (dd == bd0 && nn < bi0)) { bd1 = bd0; bi1 = bi0; bd0 = dd; bi0 = nn; }
            else                                     { bd1 = dd; bi1 = nn; }
          } else { bd2 = dd; bi2 = nn; }
        }
      }
    const int q = q0 + lane;
    const int base = (q * NSPLIT + split) * KNN;
    float* cd = ws + CAND_D;
    int*   ci = (int*)(ws + CAND_I);
    cd[base + 0] = bd0; cd[base + 1] = bd1; cd[base + 2] = bd2;
    ci[base + 0] = bi0; ci[base + 1] = bi1; ci[base + 2] = bi2;
  }
}

// =====================================================================
// Kernel 3: final merge across NSPLIT, gather Y, average, write out+idx
// =====================================================================
__global__ __launch_bounds__(128) void drknn_final(const float* __restrict__ Y,
                                                   const float* __restrict__ ws,
                                                   float* __restrict__ out) {
  int q = blockIdx.x * blockDim.x + threadIdx.x;
  if (q >= QN) return;
  const float* cd = ws + CAND_D;
  const int*   ci = (const int*)(ws + CAND_I);
  float bd0 = BIGF, bd1 = BIGF, bd2 = BIGF;
  int   bi0 = 0x7fffffff, bi1 = 0x7fffffff, bi2 = 0x7fffffff;
  for (int s = 0; s < NSPLIT * KNN; ++s) {
    float dd = cd[q * NSPLIT * KNN + s];
    int nn = ci[q * NSPLIT * KNN + s];
    if (dd < bd2 || (dd == bd2 && nn < bi2)) {
      if (dd < bd1 || (dd == bd1 && nn < bi1)) {
        bd2 = bd1; bi2 = bi1;
        if (dd < bd0 || (dd == bd0 && nn < bi0)) { bd1 = bd0; bi1 = bi0; bd0 = dd; bi0 = nn; }
        else                                     { bd1 = dd; bi1 = nn; }
      } else { bd2 = dd; bi2 = nn; }
    }
  }
#pragma unroll
  for (int j = 0; j < 3; ++j) {
    float s = Y[(size_t)bi0 * 3 + j] + Y[(size_t)bi1 * 3 + j] + Y[(size_t)bi2 * 3 + j];
    out[q * 3 + j] = s * (1.0f / 3.0f);
  }
  out[QN * 3 + q * 3 + 0] = (float)bi0;
  out[QN * 3 + q * 3 + 1] = (float)bi1;
  out[QN * 3 + q * 3 + 2] = (float)bi2;
}

// =====================================================================
extern "C" void kernel_launch(void* const* d_in, const int* in_sizes, int n_in,
                              void* d_out, int out_size, void* d_ws, size_t ws_size,
                              hipStream_t stream) {
  (void)in_sizes; (void)n_in; (void)out_size; (void)ws_size;
  const float* X  = (const float*)d_in[0];   // [N, 8, 16] -> [N,128]
  const float* Y  = (const float*)d_in[1];   // [N, 3]
  const float* Qf = (const float*)d_in[2];   // [Q, 8, 16] -> [Q,128]
  float* out = (float*)d_out;
  float* ws  = (float*)d_ws;

  drknn_norms<<<(N_PTS + 255) / 256, 256, 0, stream>>>(X, Qf, ws);
  dim3 grid(QN / 16, NSPLIT);
  drknn_main<<<grid, TPB, 0, stream>>>(X, Qf, ws);
  drknn_final<<<(QN + 127) / 128, 128, 0, stream>>>(Y, ws, out);
}